// HardNegativeMining_54133767799366
// MI455X (gfx1250) — compile-verified
//
#include <hip/hip_runtime.h>
#include <stdint.h>

// Problem constants (match reference)
#define BROWS   2048
#define NCOLS   50000
#define KSEL    101      // K = NUM_HARD_NEGATIVES + 1
#define KNEG    100      // non-positive selections per row
#define NTHREADS 1024    // 32 waves (wave32) -> fills one WGP

// ---- CDNA5 async global->LDS DMA path (guarded; falls back to plain loads) ----
#if defined(__has_builtin)
#  if __has_builtin(__builtin_amdgcn_global_load_async_to_lds_b128)
#    define USE_ASYNC_LDS 1
#  endif
#  if __has_builtin(__builtin_amdgcn_s_wait_asynccnt)
#    define HAVE_WAIT_ASYNC_BUILTIN 1
#  endif
#endif

// Builtin signature (from hipcc diagnostic): (int4 AS1*, int4 AS3*, imm, imm)
typedef int v4i __attribute__((ext_vector_type(4)));
typedef __attribute__((address_space(1))) v4i* gptr_v4i;
typedef __attribute__((address_space(3))) v4i* lptr_v4i;

// Map float -> uint32 such that unsigned compare == float compare (ascending).
__device__ __forceinline__ uint32_t sortkey(float f) {
    uint32_t u = __float_as_uint(f);
    return u ^ ((uint32_t)(((int32_t)u) >> 31) | 0x80000000u);
}

__global__ __launch_bounds__(NTHREADS, 1)
void HardNegativeMining_topk_kernel(const float* __restrict__ logits,
                                    const float* __restrict__ labels,
                                    float* __restrict__ out) {
    // 200 KB row staging -- only possible because CDNA5 has 320 KB LDS per WGP.
    __shared__ __align__(16) float s_row[NCOLS];
    __shared__ uint32_t s_hist[256];
    __shared__ uint32_t s_selidx[KNEG];
    __shared__ uint32_t s_selkey[KNEG];
    __shared__ uint32_t s_sorted[KNEG];
    __shared__ uint32_t s_prefix, s_need, s_pos, s_gt, s_eq;

    const int row = blockIdx.x;
    const int tid = threadIdx.x;
    const float* grow = logits + (size_t)row * NCOLS;
    const float* lrow = labels + (size_t)row * NCOLS;

    // ---- Stage the logits row into LDS (async DMA on CDNA5, VGPR-bypassing) ----
#ifdef USE_ASYNC_LDS
    for (int i4 = tid; i4 < NCOLS / 4; i4 += NTHREADS) {
        __builtin_amdgcn_global_load_async_to_lds_b128(
            (gptr_v4i)(grow + 4 * i4),
            (lptr_v4i)(s_row + 4 * i4),
            /*offset=*/0, /*cpol=*/0);
    }
#else
    for (int i = tid; i < NCOLS; i += NTHREADS) s_row[i] = grow[i];
#endif

    // Overlap with the single pass over labels: find the positive column.
    // Exactly one element per row is 1.0 (one-hot), so a plain store is safe.
    for (int i = tid; i < NCOLS; i += NTHREADS) {
        if (lrow[i] > 0.5f) s_pos = (uint32_t)i;
    }
    if (tid == 0) { s_prefix = 0u; s_need = KNEG; }

#ifdef USE_ASYNC_LDS
#  ifdef HAVE_WAIT_ASYNC_BUILTIN
    __builtin_amdgcn_s_wait_asynccnt(0);
#  else
    asm volatile("s_wait_asynccnt 0" ::: "memory");
#  endif
#endif
    __syncthreads();

    const uint32_t pos = s_pos;

    // ---- Exact 4x8-bit radix select (MSB -> LSB) over LDS-resident keys. ----
    // The boosted positive is always the global max, so exclude it and select
    // the top-100 of the remaining raw logits (equivalent ordering).
    for (int p = 0; p < 4; ++p) {
        for (int b = tid; b < 256; b += NTHREADS) s_hist[b] = 0u;
        __syncthreads();
        const uint32_t prefix = s_prefix;
        const int sh = 24 - 8 * p;
        for (int i = tid; i < NCOLS; i += NTHREADS) {
            if ((uint32_t)i == pos) continue;
            uint32_t key = sortkey(s_row[i]);
            if (p == 0 || (key >> (sh + 8)) == prefix)
                atomicAdd(&s_hist[(key >> sh) & 255u], 1u);
        }
        __syncthreads();
        if (tid == 0) {
            uint32_t need = s_need, cum = 0u;
            int chosen = 0;
            for (int b = 255; b >= 0; --b) {
                uint32_t c = s_hist[b];
                if (cum + c >= need) { chosen = b; break; }
                cum += c;
            }
            s_need   = need - cum;                       // still needed among == bin
            s_prefix = (prefix << 8) | (uint32_t)chosen; // extend prefix
        }
        __syncthreads();
    }

    const uint32_t T    = s_prefix;       // exact 32-bit threshold key
    const uint32_t need = s_need;         // how many == T survive
    const uint32_t G    = KNEG - need;    // count of keys strictly > T
    if (tid == 0) { s_gt = 0u; s_eq = 0u; }
    __syncthreads();

    // ---- Compaction pass: collect the 100 selected (index, key) pairs. ----
    for (int i = tid; i < NCOLS; i += NTHREADS) {
        if ((uint32_t)i == pos) continue;
        uint32_t key = sortkey(s_row[i]);
        if (key > T) {
            uint32_t slot = atomicAdd(&s_gt, 1u);        // slot < G exactly
            s_selidx[slot] = (uint32_t)i;
            s_selkey[slot] = key;
        } else if (key == T) {
            uint32_t e = atomicAdd(&s_eq, 1u);           // ties: any subset is
            uint32_t slot = G + e;                       // value-identical
            if (slot < KNEG) { s_selidx[slot] = (uint32_t)i; s_selkey[slot] = key; }
        }
    }
    __syncthreads();

    // ---- Rank sort (key desc, index asc) to match lax.top_k output order. ----
    if (tid < KNEG) {
        const uint32_t mykey = s_selkey[tid];
        const uint32_t myidx = s_selidx[tid];
        int rank = 0;
        for (int j = 0; j < KNEG; ++j) {
            uint32_t kj = s_selkey[j], ij = s_selidx[j];
            if (kj > mykey || (kj == mykey && ij < myidx)) rank++;
        }
        s_sorted[rank] = myidx;
    }
    __syncthreads();

    // ---- Emit: slot 0 is always the boosted positive; slots 1..100 sorted. ----
    float* out_logits = out;
    float* out_labels = out + (size_t)BROWS * KSEL;
    if (tid == 0) {
        out_logits[(size_t)row * KSEL] = s_row[pos];
        out_labels[(size_t)row * KSEL] = 1.0f;
    }
    if (tid < KNEG) {
        const uint32_t i = s_sorted[tid];
        out_logits[(size_t)row * KSEL + 1 + tid] = s_row[i];
        out_labels[(size_t)row * KSEL + 1 + tid] = 0.0f;
    }
}

extern "C" void kernel_launch(void* const* d_in, const int* in_sizes, int n_in,
                              void* d_out, int out_size, void* d_ws, size_t ws_size,
                              hipStream_t stream) {
    (void)in_sizes; (void)n_in; (void)out_size; (void)d_ws; (void)ws_size;
    const float* logits = (const float*)d_in[0];
    const float* labels = (const float*)d_in[1];
    float* out = (float*)d_out;
    HardNegativeMining_topk_kernel<<<dim3(BROWS), dim3(NTHREADS), 0, stream>>>(
        logits, labels, out);
}